// SimpleAttention_78280073937159
// MI455X (gfx1250) — compile-verified
//
#include <hip/hip_runtime.h>

// ---------------------------------------------------------------------------
// SimpleAttention for MI455X (gfx1250, wave32, WMMA + async-LDS copies).
// Memory-bound: K+V = 2 GiB fp32 streamed once -> ~92us floor @ 23.3 TB/s.
// Matrix math on V_WMMA_F32_16X16X4_F32 (native fp32).
// V stream staged through LDS with GLOBAL_LOAD_ASYNC_TO_LDS_B128 (ASYNCcnt),
// double buffered.
// ---------------------------------------------------------------------------

typedef __attribute__((ext_vector_type(2))) float v2f;
typedef __attribute__((ext_vector_type(8))) float v8f;
typedef int v4i __attribute__((vector_size(16)));   // 16-byte unit for async copy

#define NHEADS   32
#define HDIM     128
#define QRANK    1536
#define BATCH    16
#define SEQ      4096
#define HK       (NHEADS * HDIM)   // 4096

#define VTILE_ROWS 32              // V rows staged per async tile
#define VROW       136             // LDS row stride (floats); 272%64==16 -> no
                                   // bank conflict between WMMA B-frag halves
#define NVTILES    (SEQ / VTILE_ROWS)

#define GLOBAL_AS __attribute__((address_space(1)))
#define LDS_AS    __attribute__((address_space(3)))

// ---- CDNA5 async copy: global -> LDS, tracked on ASYNCcnt ------------------
__device__ __forceinline__ void async_load_b128(const float* g, float* l) {
#if __has_builtin(__builtin_amdgcn_global_load_async_to_lds_b128)
  __builtin_amdgcn_global_load_async_to_lds_b128(
      (GLOBAL_AS v4i*)(float*)g, (LDS_AS v4i*)l, 0, 0);
#else
  unsigned lds_off = (unsigned)(__UINTPTR_TYPE__)(LDS_AS void*)l;
  asm volatile("global_load_async_to_lds_b128 %0, %1, off"
               :: "v"(lds_off), "v"(g) : "memory");
#endif
}

#if __has_builtin(__builtin_amdgcn_s_wait_asynccnt)
#define WAIT_ASYNC(N) __builtin_amdgcn_s_wait_asynccnt(N)
#else
#define WAIT_ASYNC(N) asm volatile("s_wait_asynccnt %0" :: "i"(N))
#endif

// ---------------------------------------------------------------------------
// WMMA fragment conventions (wave32):
//   A (16x4 fp32, ISA 7.12.2): lane&15 = M row; lane>>4 = K half; VGPR0/1 =
//     K{0,1} (lanes 0-15) / K{2,3} (lanes 16-31) -> one float2 per lane.
//   B (4x16 fp32, mirror-of-A): lane&15 = N col; lane>>4 = K half.
//   C/D (16x16 fp32, documented): VGPR v = M=v (lanes 0-15) / M=v+8
//     (lanes 16-31), N = lane&15.
// ---------------------------------------------------------------------------

// Kernel 1: q[b,n] = hsq[b,:] . w_q[n,:] + b_q[n]  (16 x 4096 x 1536 GEMM)
// M = batch = 16 -> full 16x16 tile utilization. One wave per 16-wide N tile.
__global__ __launch_bounds__(128) void qproj_wmma(
    const float* __restrict__ hsq, const float* __restrict__ w_q,
    const float* __restrict__ b_q, float* __restrict__ q_out)
{
  const int lane = threadIdx.x & 31;
  const int wave = threadIdx.x >> 5;
  const int n0   = (blockIdx.x * 4 + wave) * 16;
  const int m    = lane & 15;
  const int hs   = lane >> 4;

  const float* arow = hsq + (size_t)m * QRANK + 2 * hs;
  const float* brow = w_q + (size_t)(n0 + m) * QRANK + 2 * hs;

  v8f acc = {};
  #pragma unroll 8
  for (int c0 = 0; c0 < QRANK; c0 += 4) {
    v2f a = *(const v2f*)(arow + c0);
    v2f b = *(const v2f*)(brow + c0);
    acc = __builtin_amdgcn_wmma_f32_16x16x4_f32(false, a, false, b,
                                                (short)0, acc, false, false);
  }
  const float bias = b_q[n0 + m];
  #pragma unroll
  for (int v = 0; v < 8; ++v)
    q_out[(size_t)(v + 8 * hs) * HK + n0 + m] = acc[v] + bias;
}

// Kernel 2: one block (256 threads / 8 waves) per (b, h).
__global__ __launch_bounds__(256) void attn_wmma(
    const float* __restrict__ key, const float* __restrict__ val,
    float* __restrict__ qio)   // in: staged q; out: attention output
{
  __shared__ float s_scores[SEQ];                  // 16 KB
  __shared__ float s_q[HDIM];
  __shared__ float s_red[16];
  __shared__ float s_v[2][VTILE_ROWS * VROW];      // 2 x 17 KB async V staging

  const int tid  = threadIdx.x;
  const int lane = tid & 31;
  const int wave = tid >> 5;
  const int h    = blockIdx.x;
  const int b    = blockIdx.y;
  const int m    = lane & 15;
  const int hs   = lane >> 4;

  float* slice = qio + (size_t)b * HK + h * HDIM;  // this block's q / out slice
  if (tid < HDIM) s_q[tid] = slice[tid];
  __syncthreads();

  // ---- Phase A: scores[l] = q . K[b,l,h,:]  (WMMA, q broadcast in B) ----
  const float* Kb = key + (size_t)b * SEQ * HK + h * HDIM;
  for (int t = wave; t < SEQ / 16; t += 8) {
    const int l0 = t * 16;
    const float* arow = Kb + (size_t)(l0 + m) * HK + 2 * hs;
    __builtin_prefetch(Kb + (size_t)(l0 + 128 + m) * HK + 2 * hs, 0, 1);
    v8f acc = {};
    #pragma unroll
    for (int k0 = 0; k0 < HDIM; k0 += 4) {
      v2f a  = *(const v2f*)(arow + k0);
      v2f bq = *(const v2f*)(&s_q[k0 + 2 * hs]);
      acc = __builtin_amdgcn_wmma_f32_16x16x4_f32(false, a, false, bq,
                                                  (short)0, acc, false, false);
    }
    if (m == 0) {                       // every C column equal -> harvest N=0
      #pragma unroll
      for (int v = 0; v < 8; ++v) s_scores[l0 + v + 8 * hs] = acc[v];
    }
  }
  __syncthreads();

  // ---- Phase B: softmax (1/sum folded into final store) ----------------
  float lmax = -3.402823466e38f;
  for (int i = tid; i < SEQ; i += 256) lmax = fmaxf(lmax, s_scores[i]);
  #pragma unroll
  for (int off = 16; off > 0; off >>= 1)
    lmax = fmaxf(lmax, __shfl_xor(lmax, off, 32));
  if (lane == 0) s_red[wave] = lmax;
  __syncthreads();
  float gmax = s_red[0];
  #pragma unroll
  for (int w = 1; w < 8; ++w) gmax = fmaxf(gmax, s_red[w]);

  float lsum = 0.f;
  for (int i = tid; i < SEQ; i += 256) {
    float e = __expf(s_scores[i] - gmax);
    s_scores[i] = e;
    lsum += e;
  }
  #pragma unroll
  for (int off = 16; off > 0; off >>= 1) lsum += __shfl_xor(lsum, off, 32);
  if (lane == 0) s_red[8 + wave] = lsum;
  __syncthreads();
  float tot = 0.f;
  #pragma unroll
  for (int w = 0; w < 8; ++w) tot += s_red[8 + w];
  const float inv = 1.0f / tot;

  // ---- Phase C: O[k] = sum_l p[l] V[b,l,h,k] ---------------------------
  // V streamed through LDS via async b128 copies, double buffered.
  // Stage: thread tid -> tile row (tid>>3), 16-float segment (tid&7):
  //        8 lanes copy 16B each at 64B stride (128B of a 512B row); 1 async instr/wave.
  const float* Vh = val + (size_t)b * SEQ * HK + h * HDIM;
  const int r_st   = tid >> 3;
  const int seg_st = (tid & 7) * 16;

  // prologue: stage tile 0
  async_load_b128(Vh + (size_t)r_st * HK + seg_st,
                  &s_v[0][r_st * VROW + seg_st]);

  const int kb = wave * 16;       // this wave's head-dim tile
  v8f acc = {};
  for (int t = 0; t < NVTILES; ++t) {
    if (t + 1 < NVTILES) {
      async_load_b128(Vh + (size_t)((t + 1) * VTILE_ROWS + r_st) * HK + seg_st,
                      &s_v[(t + 1) & 1][r_st * VROW + seg_st]);
      WAIT_ASYNC(1);              // tile t complete (only newest outstanding)
    } else {
      WAIT_ASYNC(0);
    }
    __syncthreads();              // all waves' async writes visible

    const float* buf = s_v[t & 1];
    const float* ps  = &s_scores[t * VTILE_ROWS];
    #pragma unroll
    for (int l0 = 0; l0 < VTILE_ROWS; l0 += 4) {
      v2f a = *(const v2f*)(ps + l0 + 2 * hs);          // A rows = p broadcast
      v2f bv;
      bv.x = buf[(l0 + 2 * hs) * VROW + kb + m];        // B[k,n]=V[l0+k, kb+n]
      bv.y = buf[(l0 + 2 * hs + 1) * VROW + kb + m];
      acc = __builtin_amdgcn_wmma_f32_16x16x4_f32(false, a, false, bv,
                                                  (short)0, acc, false, false);
    }
    __syncthreads();              // reads done before buffer reuse (t+2)
  }
  // every C row equal -> row M=0 in acc[0], lanes 0-15 (N = lane)
  if (lane < 16) slice[kb + lane] = acc[0] * inv;
}

// ---------------------------------------------------------------------------
extern "C" void kernel_launch(void* const* d_in, const int* in_sizes, int n_in,
                              void* d_out, int out_size, void* d_ws, size_t ws_size,
                              hipStream_t stream) {
  (void)in_sizes; (void)n_in; (void)out_size; (void)d_ws; (void)ws_size;
  const float* hsq = (const float*)d_in[0];   // [16, 1536]
  const float* key = (const float*)d_in[1];   // [16, 4096, 32, 128]
  const float* val = (const float*)d_in[2];   // [16, 4096, 32, 128]
  const float* w_q = (const float*)d_in[3];   // [4096, 1536]
  const float* b_q = (const float*)d_in[4];   // [4096]
  float* out = (float*)d_out;                 // [16, 4096]

  // Stage q into d_out; each attn block reads only its own slice then
  // overwrites it -> deterministic, no inter-block aliasing.
  qproj_wmma<<<dim3(64), dim3(128), 0, stream>>>(hsq, w_q, b_q, out);
  attn_wmma<<<dim3(NHEADS, BATCH), dim3(256), 0, stream>>>(key, val, out);
}